// Colorcal_6536940224717
// MI455X (gfx1250) — compile-verified
//
#include <hip/hip_runtime.h>
#include <hip/hip_bf16.h>

// Per-camera per-channel affine (groups=3 1x1 conv == scale+bias).
// Memory-bound: 402 MB HBM traffic, ~17us floor at 23.3 TB/s on MI455X.
// Strategy: 128-bit nontemporal loads/stores, scalar (SGPR) coefficient
// fetch, 4 in-flight vec4s per lane, wave32 256-thread blocks.

typedef float v4f __attribute__((ext_vector_type(4)));

#define PLANE_ELEMS (1024 * 1024)   // floats per [channel] plane
#define BLOCK_THREADS 256
#define VEC4_PER_THREAD 4
#define BLOCKS_X (PLANE_ELEMS / 4 / BLOCK_THREADS / VEC4_PER_THREAD)  // 256

__global__ __launch_bounds__(BLOCK_THREADS)
void Colorcal_6536940224717_kernel(const float* __restrict__ img,
                                   const int* __restrict__ camindex,
                                   const float* __restrict__ weight,
                                   const float* __restrict__ bias,
                                   float* __restrict__ out)
{
    // plane = b*3 + c, one (scale, bias) pair per plane -> uniform per block,
    // so these become scalar (s_load) fetches, no per-lane VMEM.
    const int plane = blockIdx.y;        // 0..47
    const int b     = plane / 3;
    const int c     = plane - 3 * b;
    const int cam   = camindex[b];
    const float w   = weight[cam * 3 + c];
    const float bi  = bias[cam * 3 + c];

    const size_t planeBase = (size_t)plane * PLANE_ELEMS;
    const v4f* __restrict__ src = reinterpret_cast<const v4f*>(img + planeBase);
    v4f* __restrict__ dst       = reinterpret_cast<v4f*>(out + planeBase);

    const int base   = blockIdx.x * BLOCK_THREADS + threadIdx.x;
    const int stride = BLOCKS_X * BLOCK_THREADS;   // vec4 stride between the 4 chunks

    // Issue all 4 nontemporal b128 loads first (4 outstanding on LOADcnt),
    // then FMA + nontemporal b128 stores. Streaming data: NT hint keeps the
    // 402MB stream from thrashing WGP$/L2 (touched exactly once).
    v4f v[VEC4_PER_THREAD];
#pragma unroll
    for (int u = 0; u < VEC4_PER_THREAD; ++u) {
        v[u] = __builtin_nontemporal_load(src + base + u * stride);
    }
#pragma unroll
    for (int u = 0; u < VEC4_PER_THREAD; ++u) {
        v4f r;
        r.x = fmaf(v[u].x, w, bi);
        r.y = fmaf(v[u].y, w, bi);
        r.z = fmaf(v[u].z, w, bi);
        r.w = fmaf(v[u].w, w, bi);
        __builtin_nontemporal_store(r, dst + base + u * stride);
    }
}

extern "C" void kernel_launch(void* const* d_in, const int* in_sizes, int n_in,
                              void* d_out, int out_size, void* d_ws, size_t ws_size,
                              hipStream_t stream) {
    const float* img      = (const float*)d_in[0];   // [16,3,1024,1024] f32
    const int*   camindex = (const int*)d_in[1];     // [16] int
    const float* weight   = (const float*)d_in[2];   // [16,3] f32
    const float* bias     = (const float*)d_in[3];   // [16,3] f32
    float* out            = (float*)d_out;           // [16,3,1024,1024] f32

    dim3 grid(BLOCKS_X, 16 * 3);     // (256, 48): gridDim.y = plane index
    dim3 block(BLOCK_THREADS);       // 256 threads = 8 wave32 waves
    Colorcal_6536940224717_kernel<<<grid, block, 0, stream>>>(
        img, camindex, weight, bias, out);
}